// Net_88356067213404
// MI455X (gfx1250) — compile-verified
//
#include <hip/hip_runtime.h>

// Problem constants (from the reference)
#define NBASIS   25
#define DOF      2
#define D_IN     54              // DOF * (NBASIS + 2)
#define T_STEPS  301
#define TAU      3.0f
#define DT       0.01f
#define A_X      2.0f
#define A_Z      48.0f
#define B_Z      12.0f           // A_Z / 4

// Tiling
#define ROWS     64              // batch rows per block
#define THREADS  128             // 4 wave32 per block
#define K_PAD    56              // K padded to multiple of 4 (14 k-steps)
#define LDX      57              // odd LDS stride -> no bank conflicts
#define LDW      57

typedef __attribute__((ext_vector_type(2))) float v2f;
typedef __attribute__((ext_vector_type(8))) float v8f;

__global__ __launch_bounds__(THREADS)
void dmp_fused_kernel(const float* __restrict__ x,   const float* __restrict__ Wg,
                      const float* __restrict__ bvec, const float* __restrict__ cvec,
                      const float* __restrict__ s2vec, const float* __restrict__ svec,
                      float* __restrict__ out, int Bsz)
{
    // LDS: 14592 + 14592 + 30100 = 59284 bytes
    __shared__ float xs[ROWS * LDX];          // x tile; reused as h tile after GEMM
    __shared__ float Ws[64 * LDW];            // zero-padded W (rows 54..63 = 0)
    __shared__ float psih[T_STEPS * NBASIS];  // psihat[t][n] = psi*xp/sum(psi)

    const int tid  = threadIdx.x;
    const int row0 = blockIdx.x * ROWS;

    // ---- zero-fill padded staging buffers ----
    for (int i = tid; i < ROWS * LDX; i += THREADS) xs[i] = 0.0f;
    for (int i = tid; i < 64   * LDW; i += THREADS) Ws[i] = 0.0f;
    __syncthreads();

    // ---- stage x tile ----
    for (int i = tid; i < ROWS * D_IN; i += THREADS) {
        int r = i / D_IN, k = i - r * D_IN;
        int rg = row0 + r;
        xs[r * LDX + k] = (rg < Bsz) ? x[(size_t)rg * D_IN + k] : 0.0f;
    }
    // ---- stage W (row-major, B-matrix of h = x @ W^T) ----
    for (int i = tid; i < D_IN * D_IN; i += THREADS) {
        int n = i / D_IN, k = i - n * D_IN;
        Ws[n * LDW + k] = Wg[i];
    }
    // ---- batch-invariant basis table: psihat[t][n] = psi(t,n) * xp_t / sum_n psi ----
    const float logr = __logf(1.0f - A_X * DT / TAU);  // xp_t = r^t (Euler phase)
    for (int t = tid; t < T_STEPS; t += THREADS) {
        float xp = __expf((float)t * logr);
        float psi[NBASIS];
        float S = 0.0f;
        #pragma unroll
        for (int n = 0; n < NBASIS; ++n) {
            float d = xp - cvec[n];
            psi[n] = __expf(-0.5f * d * d / s2vec[n]);
            S += psi[n];
        }
        float m = xp / S;
        #pragma unroll
        for (int n = 0; n < NBASIS; ++n) psih[t * NBASIS + n] = psi[n] * m;
    }
    __syncthreads();

    // ---- WMMA f32 16x16x4 GEMM: 64x64 block tile, K = 56 (zero padded) ----
    const int wave = tid >> 5;
    const int lane = tid & 31;
    const int l    = lane & 15;
    const int hi   = lane >> 4;          // 0: K{0,1} / N lanes 0-15 ; 1: K{2,3}
    const int m0   = wave * 16;

    v8f acc[4] = {};                     // 4 N-tiles of 16, f32 accumulators
    for (int k0 = 0; k0 < K_PAD; k0 += 4) {
        v2f a;                           // A 16x4: lane(l,hi) -> row m0+l, K = k0+2*hi{,+1}
        a.x = xs[(m0 + l) * LDX + k0 + 2 * hi];
        a.y = xs[(m0 + l) * LDX + k0 + 2 * hi + 1];
        #pragma unroll
        for (int nt = 0; nt < 4; ++nt) {
            int n0 = nt * 16;            // B 4x16 = W^T: lane(l,hi) -> col n0+l, K = k0+2*hi{,+1}
            v2f bm;
            bm.x = Ws[(n0 + l) * LDW + k0 + 2 * hi];
            bm.y = Ws[(n0 + l) * LDW + k0 + 2 * hi + 1];
            acc[nt] = __builtin_amdgcn_wmma_f32_16x16x4_f32(
                          false, a, false, bm, (short)0, acc[nt], false, false);
        }
    }
    __syncthreads();   // everyone done reading xs; safe to reuse as h tile

    // ---- scatter D to LDS with bias+scale: p = (h + b) * scale ----
    // D layout: vgpr r -> M = m0 + r + 8*hi, N = n0 + l
    #pragma unroll
    for (int nt = 0; nt < 4; ++nt) {
        int n = nt * 16 + l;
        if (n < D_IN) {
            float bb = bvec[n], ss = svec[n];
            #pragma unroll
            for (int r = 0; r < 8; ++r) {
                xs[(m0 + r + 8 * hi) * LDX + n] = (acc[nt][r] + bb) * ss;
            }
        }
    }
    __syncthreads();

    // ---- per-(row,dof) DMP recurrence, 301 steps ----
    const int rr = tid >> 1;
    const int dd = tid & 1;
    const int rg = row0 + rr;
    if (rg < Bsz) {
        const float* p = &xs[rr * LDX + dd * (NBASIS + 2)];
        const float y0 = p[0];
        const float g  = p[1];
        float w[NBASIS];
        #pragma unroll
        for (int n = 0; n < NBASIS; ++n) w[n] = p[2 + n];

        float y = y0, z = 0.0f;
        const float gmy0 = g - y0;
        const float k1   = DT / TAU;
        float* o = out + (size_t)rg * (DOF * T_STEPS) + (size_t)dd * T_STEPS;

        for (int t = 0; t < T_STEPS; ++t) {
            o[t] = y;                                   // scan emits pre-update y
            const float* ph = &psih[t * NBASIS];        // wave-uniform LDS broadcast
            float fx = 0.0f;
            #pragma unroll
            for (int n = 0; n < NBASIS; ++n) fx += w[n] * ph[n];
            fx *= gmy0;
            float dz = A_Z * (B_Z * (g - y) - z) + fx;  // (…)/TAU folded into k1
            float yn = y + z * k1;                      // uses old z
            z += dz * k1;
            y  = yn;
        }
    }
}

extern "C" void kernel_launch(void* const* d_in, const int* in_sizes, int n_in,
                              void* d_out, int out_size, void* d_ws, size_t ws_size,
                              hipStream_t stream) {
    const float* x  = (const float*)d_in[0];
    const float* W  = (const float*)d_in[1];
    const float* b  = (const float*)d_in[2];
    const float* c  = (const float*)d_in[3];
    const float* s2 = (const float*)d_in[4];
    const float* sc = (const float*)d_in[5];
    const int Bsz = in_sizes[0] / D_IN;
    const int grid = (Bsz + ROWS - 1) / ROWS;
    dmp_fused_kernel<<<grid, THREADS, 0, stream>>>(x, W, b, c, s2, sc,
                                                   (float*)d_out, Bsz);
}